// DPNResBlock_31774168055982
// MI455X (gfx1250) — compile-verified
//
#include <hip/hip_runtime.h>
#include <hip/hip_bf16.h>
#include <math.h>

// ---------------- problem constants ----------------
#define BB 4
#define CC 64
#define HH 64
#define WW 256
#define OO 64
#define KK 3
#define PP 9          // K*K
#define HP 66         // H + 2
#define WP 258        // W + 2
#define HPWP (HP*WP)  // 17028
#define CP 576        // C*P, GEMM K dim
#define NPIX (HH*WW)  // 16384 per batch image
#define NY (BB*OO*HH*WW)          // 4,194,304 elements of y
#define OH 31
#define OW 127
#define NOUT (BB*128*OH*OW)       // 2,015,744

// workspace layout (float offsets)
#define XP_OFF   0u
#define XP_N     ((size_t)BB*CC*HPWP)              // 4,359,168
#define OFFS_OFF (XP_OFF + XP_N)
#define OFFS_N   ((size_t)BB*18*HH*WW)             // 1,179,648
#define Y_OFF    (OFFS_OFF + OFFS_N)
#define Y_N      ((size_t)NY)
#define WA_OFF   (Y_OFF + Y_N)
#define WA_N     ((size_t)4*144*32*2)              // 36,864
#define ST_OFF   (WA_OFF + WA_N)                   // 2 floats

typedef __attribute__((ext_vector_type(2))) float v2f;
typedef __attribute__((ext_vector_type(8))) float v8f;

// ---------------- helpers ----------------
__device__ __forceinline__ float triwave(float x) {
    const float PI = 3.14159265358979323846f;
    float n = floorf(x / PI + 0.5f);
    float m = n - 2.0f * floorf(n * 0.5f);     // mod(n,2) in [0,2)
    float sgn = 1.0f - 2.0f * m;
    return (x - PI * n) * sgn;
}
__device__ __forceinline__ float prak_f(float x) {
    return triwave(x) + triwave(x + 1.57079632679489662f);
}

// ---------------- kernel 0: zero stats ----------------
__global__ void init_stats_kernel(float* stats) {
    if (threadIdx.x < 2) stats[threadIdx.x] = 0.0f;
}

// ---------------- kernel 1: pad x -> xp ----------------
__global__ void pad_kernel(const float* __restrict__ x, float* __restrict__ xp) {
    size_t gid = (size_t)blockIdx.x * blockDim.x + threadIdx.x;
    if (gid >= XP_N) return;
    int wp = (int)(gid % WP);
    size_t t = gid / WP;
    int hp = (int)(t % HP);
    size_t bc = t / HP;                       // b*C + c
    float v = 0.0f;
    int h = hp - 1, w = wp - 1;
    if (h >= 0 && h < HH && w >= 0 && w < WW)
        v = x[bc * (size_t)(HH*WW) + (size_t)h * WW + w];
    xp[gid] = v;
}

// ---------------- kernel 2: swizzle weight into WMMA A-fragment order ----------------
// wA[((m*144+s)*32+lane)*2 + {0,1}] = W[16m + lane%16][4s + (lane<16?0:2) + {0,1}]
__global__ void prep_wA_kernel(const float* __restrict__ weight, float* __restrict__ wA) {
    int gid = blockIdx.x * blockDim.x + threadIdx.x;
    if (gid >= 4*144*32) return;
    int lane = gid & 31;
    int s = (gid >> 5) % 144;
    int m = gid / (144*32);
    int koff = (lane < 16) ? 0 : 2;
    int o = 16*m + (lane & 15);
    int k = 4*s + koff;
    wA[(size_t)gid*2 + 0] = weight[(size_t)o*CP + k];
    wA[(size_t)gid*2 + 1] = weight[(size_t)o*CP + k + 1];
}

// ---------------- kernel 3: offset conv (3x3, C=64 -> 18) ----------------
__global__ void offset_conv_kernel(const float* __restrict__ xp,
                                   const float* __restrict__ ow,
                                   const float* __restrict__ ob,
                                   float* __restrict__ offs) {
    __shared__ float s_w[18*CP];   // 41,472 B
    __shared__ float s_b[18];
    for (int i = threadIdx.x; i < 18*CP; i += blockDim.x) s_w[i] = ow[i];
    if (threadIdx.x < 18) s_b[threadIdx.x] = ob[threadIdx.x];
    __syncthreads();

    size_t gid = (size_t)blockIdx.x * blockDim.x + threadIdx.x;
    if (gid >= OFFS_N) return;
    int w = (int)(gid % WW);
    size_t t = gid / WW;
    int h = (int)(t % HH);  t /= HH;
    int oc = (int)(t % 18);
    int b  = (int)(t / 18);

    const float* xb = xp + (size_t)b * CC * HPWP;
    const float* wrow = s_w + oc * CP;
    float acc = s_b[oc];
    for (int c = 0; c < CC; ++c) {
        const float* xc = xb + (size_t)c * HPWP;
        #pragma unroll
        for (int i = 0; i < 3; ++i) {
            const float* xr = xc + (size_t)(h + i) * WP + w;
            #pragma unroll
            for (int j = 0; j < 3; ++j)
                acc = fmaf(xr[j], wrow[c*9 + i*3 + j], acc);
        }
    }
    offs[gid] = acc;
}

// ---------------- kernel 4: deformable sampling + WMMA GEMM ----------------
// grid: B*H*(W/16) blocks, 128 threads (4 waves). Wave m owns output channels [16m,16m+16).
__global__ void deform_wmma_kernel(const float* __restrict__ xp,
                                   const float* __restrict__ offs,
                                   const float* __restrict__ mask,
                                   const float* __restrict__ wA,
                                   float* __restrict__ y) {
    __shared__ float s_val[CP*16];     // 36,864 B : val[k][n], row stride 16
    __shared__ int   s_gidx[144*4];    // per (p,n): 4 corner indices
    __shared__ float s_gw[144*4];      // per (p,n): 4 weights (premul by mask)

    const int tid  = threadIdx.x;
    const int lane = tid & 31;
    const int wv   = tid >> 5;

    int blk = blockIdx.x;
    const int WT = WW / 16;            // 16 tiles along W
    int b  = blk / (HH * WT);
    int r0 = blk % (HH * WT);
    int h  = r0 / WT;
    int w0 = (r0 % WT) * 16;

    // ---- stage 1: per (p, n) bilinear coords/weights ----
    for (int r = tid; r < 144; r += blockDim.x) {
        int p = r >> 4;                // 0..8
        int n = r & 15;
        int w = w0 + n;
        float oy = offs[(((size_t)b*18 + 2*p    )*HH + h)*WW + w];
        float ox = offs[(((size_t)b*18 + 2*p + 1)*HH + h)*WW + w];
        float mk = mask[(((size_t)b*PP + p)*HH + h)*WW + w];
        float py = (float)(h + p/3) + oy;
        float px = (float)(w + p%3) + ox;
        float y0 = floorf(py), x0 = floorf(px);
        float wy1 = py - y0, wx1 = px - x0;
        float wy0 = 1.0f - wy1, wx0 = 1.0f - wx1;
        #pragma unroll
        for (int cidx = 0; cidx < 4; ++cidx) {
            float yc = y0 + (float)(cidx >> 1);
            float xc = x0 + (float)(cidx & 1);
            float valid = (yc >= 0.0f && yc <= (float)(HP-1) &&
                           xc >= 0.0f && xc <= (float)(WP-1)) ? 1.0f : 0.0f;
            int yi = (int)fminf(fmaxf(yc, 0.0f), (float)(HP-1));
            int xi = (int)fminf(fmaxf(xc, 0.0f), (float)(WP-1));
            float wy = (cidx >> 1) ? wy1 : wy0;
            float wx = (cidx & 1)  ? wx1 : wx0;
            s_gidx[r*4 + cidx] = yi * WP + xi;
            s_gw  [r*4 + cidx] = wy * wx * valid * mk;
        }
    }
    __syncthreads();

    // ---- stage 2: gather val[c*9+p][n] into LDS ----
    const float* xb = xp + (size_t)b * CC * HPWP;
    for (int e = tid; e < CP*16; e += blockDim.x) {
        int n   = e & 15;
        int row = e >> 4;              // c*9 + p
        int c   = row / 9;
        int p   = row - 9*c;
        int pn  = (p << 4) | n;
        const float* xc = xb + (size_t)c * HPWP;
        float v = 0.0f;
        #pragma unroll
        for (int cidx = 0; cidx < 4; ++cidx)
            v = fmaf(xc[s_gidx[pn*4 + cidx]], s_gw[pn*4 + cidx], v);
        s_val[row*16 + n] = v;
    }
    __syncthreads();

    // ---- stage 3: GEMM  D[16x16] = A(weight tile) x B(val tile) ----
    const v2f* wAv = (const v2f*)wA;
    v8f acc = {};
    int nn   = lane & 15;
    int koff = (lane < 16) ? 0 : 2;
    for (int s = 0; s < 144; ++s) {
        v2f a = wAv[((size_t)wv*144 + s)*32 + lane];
        v2f bfrag;
        int k0 = 4*s + koff;
        bfrag.x = s_val[(k0    )*16 + nn];
        bfrag.y = s_val[(k0 + 1)*16 + nn];
        acc = __builtin_amdgcn_wmma_f32_16x16x4_f32(
                  false, a, false, bfrag, (short)0, acc, false, false);
    }

    // ---- stage 4: store D into y ----
    int mhi = (lane < 16) ? 0 : 8;
    #pragma unroll
    for (int r = 0; r < 8; ++r) {
        int o = 16*wv + r + mhi;
        y[(((size_t)b*OO + o)*HH + h)*WW + w0 + nn] = acc[r];
    }
}

// ---------------- kernel 5: global sum / sumsq of y ----------------
__global__ void reduce_kernel(const float* __restrict__ y, float* stats) {
    __shared__ float s1[256];
    __shared__ float s2[256];
    float a = 0.0f, bsum = 0.0f;
    for (size_t i = (size_t)blockIdx.x * blockDim.x + threadIdx.x; i < (size_t)NY;
         i += (size_t)gridDim.x * blockDim.x) {
        float v = y[i];
        a += v; bsum += v*v;
    }
    s1[threadIdx.x] = a; s2[threadIdx.x] = bsum;
    __syncthreads();
    for (int off = 128; off > 0; off >>= 1) {
        if (threadIdx.x < off) {
            s1[threadIdx.x] += s1[threadIdx.x + off];
            s2[threadIdx.x] += s2[threadIdx.x + off];
        }
        __syncthreads();
    }
    if (threadIdx.x == 0) {
        atomicAdd(&stats[0], s1[0]);
        atomicAdd(&stats[1], s2[0]);
    }
}

// ---------------- kernel 6: normalize + prak + concat + 3x3/2 maxpool ----------------
__global__ void final_kernel(const float* __restrict__ y, const float* __restrict__ x,
                             const float* __restrict__ stats, float* __restrict__ out) {
    size_t gid = (size_t)blockIdx.x * blockDim.x + threadIdx.x;
    if (gid >= (size_t)NOUT) return;
    int ow = (int)(gid % OW);
    size_t t = gid / OW;
    int oh = (int)(t % OH); t /= OH;
    int oc = (int)(t % 128);
    int b  = (int)(t / 128);

    float mu   = stats[0] / (float)NY;
    float var  = stats[1] / (float)NY - mu*mu;
    float rinv = rsqrtf(var + 1e-5f);

    int h0 = 2*oh, w0 = 2*ow;
    float m = -INFINITY;
    if (oc < 64) {
        const float* yr = y + (((size_t)b*OO + oc)*HH + h0)*WW + w0;
        #pragma unroll
        for (int i = 0; i < 3; ++i)
            #pragma unroll
            for (int j = 0; j < 3; ++j) {
                float v = prak_f((yr[(size_t)i*WW + j] - mu) * rinv);
                m = fmaxf(m, v);
            }
    } else {
        const float* xr = x + (((size_t)b*CC + (oc-64))*HH + h0)*WW + w0;
        #pragma unroll
        for (int i = 0; i < 3; ++i)
            #pragma unroll
            for (int j = 0; j < 3; ++j)
                m = fmaxf(m, xr[(size_t)i*WW + j]);
    }
    out[gid] = m;
}

// ---------------- launcher ----------------
extern "C" void kernel_launch(void* const* d_in, const int* in_sizes, int n_in,
                              void* d_out, int out_size, void* d_ws, size_t ws_size,
                              hipStream_t stream) {
    const float* x        = (const float*)d_in[0];
    const float* offset_w = (const float*)d_in[1];
    const float* offset_b = (const float*)d_in[2];
    const float* weight   = (const float*)d_in[3];
    const float* mask     = (const float*)d_in[4];
    float* out = (float*)d_out;

    float* ws    = (float*)d_ws;
    float* xp    = ws + XP_OFF;
    float* offs  = ws + OFFS_OFF;
    float* yb    = ws + Y_OFF;
    float* wA    = ws + WA_OFF;
    float* stats = ws + ST_OFF;

    init_stats_kernel<<<1, 32, 0, stream>>>(stats);

    {   size_t n = XP_N;
        pad_kernel<<<(unsigned)((n + 255) / 256), 256, 0, stream>>>(x, xp); }

    prep_wA_kernel<<<(4*144*32 + 255) / 256, 256, 0, stream>>>(weight, wA);

    {   size_t n = OFFS_N;
        offset_conv_kernel<<<(unsigned)((n + 255) / 256), 256, 0, stream>>>(xp, offset_w, offset_b, offs); }

    {   int nblk = BB * HH * (WW / 16);   // 4096
        deform_wmma_kernel<<<nblk, 128, 0, stream>>>(xp, offs, mask, wA, yb); }

    reduce_kernel<<<1024, 256, 0, stream>>>(yb, stats);

    final_kernel<<<(NOUT + 255) / 256, 256, 0, stream>>>(yb, x, stats, out);
}